// NDat_43336220016699
// MI455X (gfx1250) — compile-verified
//
#include <hip/hip_runtime.h>
#include <math.h>

#define B_ 16
#define C_ 128
#define H_ 28
#define W_ 28
#define HW_ 784
#define T_ 12544          // B*H*W
#define NH_ 4
#define HD_ 32
#define KW_ 3
#define G_ 2
#define CG_ 64
#define KS_ 5
#define OC_ 256           // 2*C
#define OH_ 14
#define OW_ 14
#define OP_ 196           // OH*OW
#define KCONV_ 1152       // C*9
#define EPS_ 1e-5f
#define LDS_PITCH 132     // 128 + 4 floats padding -> conflict-free ds_load_b64

typedef float v2f __attribute__((ext_vector_type(2)));
typedef float v8f __attribute__((ext_vector_type(8)));

__device__ __forceinline__ float wave_sum(float v) {
  #pragma unroll
  for (int m = 16; m >= 1; m >>= 1) v += __shfl_xor(v, m, 32);
  return v;
}
__device__ __forceinline__ float wave_max(float v) {
  #pragma unroll
  for (int m = 16; m >= 1; m >>= 1) v = fmaxf(v, __shfl_xor(v, m, 32));
  return v;
}

// ---------------- K1: NCHW -> token-major (t, c) ----------------
__global__ __launch_bounds__(256) void k_nchw_to_tok(const float* __restrict__ x,
                                                     float* __restrict__ xl) {
  int tid = blockIdx.x * blockDim.x + threadIdx.x;     // over B*C*HW
  if (tid >= B_ * C_ * HW_) return;
  int n = tid % HW_;
  int c = (tid / HW_) % C_;
  int b = tid / (HW_ * C_);
  xl[(size_t)(b * HW_ + n) * C_ + c] = x[tid];
}

// ---- WMMA token GEMM with async-LDS weight staging --------------------------
// Y[T,N] = X[T,128] * W[N,128]^T + bias.
// Block = 8 waves; block covers 128 rows (one m-tile per wave) x 64 cols.
// The 64x128 weight slab is async-DMA'd into LDS once per block
// (GLOBAL_LOAD_ASYNC_TO_LDS_B128 / ASYNCcnt) and shared by all waves.
// Each wave holds 4 accumulating 16x16 WMMA tiles -> A fetched once per k-step.
__global__ __launch_bounds__(256) void k_gemm_wmma_lds(const float* __restrict__ X,
                                                       const float* __restrict__ Wt,
                                                       const float* __restrict__ bias,
                                                       float* __restrict__ Y,
                                                       int N) {
  __shared__ float sw[64 * LDS_PITCH];                 // 33.8 KB of 320 KB/WGP
  const int lane = threadIdx.x & 31;
  const int wv   = threadIdx.x >> 5;
  const int n0   = blockIdx.y * 64;

  // ---- stage W[n0 .. n0+64) x 128 into LDS via async global->LDS DMA ----
  {
    const int t = threadIdx.x;                         // 2048 16B-chunks total
    #pragma unroll
    for (int i = 0; i < 8; ++i) {
      int chunk = i * 256 + t;
      int row   = chunk >> 5;                          // 32 chunks per 128-f row
      int col4  = (chunk & 31) << 2;
      unsigned lds_addr =
          (unsigned)(unsigned long long)&sw[row * LDS_PITCH + col4];
      unsigned long long gaddr =
          (unsigned long long)(Wt + (size_t)(n0 + row) * 128 + col4);
      asm volatile("global_load_async_to_lds_b128 %0, %1, off"
                   :: "v"(lds_addr), "v"(gaddr)
                   : "memory");
    }
    asm volatile("s_wait_asynccnt 0x0" ::: "memory");
  }
  __syncthreads();

  const int mt = blockIdx.x * 8 + wv;                  // m-tile (0..783)
  const int r  = lane & 15;
  const int kb = (lane >> 4) << 1;                     // 0 or 2
  const float* xrow = X + (size_t)(mt * 16 + r) * 128;

  v8f acc[4] = {};
  for (int k = 0; k < 128; k += 4) {
    v2f a = *(const v2f*)(xrow + k + kb);
    #pragma unroll
    for (int nt = 0; nt < 4; ++nt) {
      v2f b = *(const v2f*)(&sw[(nt * 16 + r) * LDS_PITCH + k + kb]);
      acc[nt] = __builtin_amdgcn_wmma_f32_16x16x4_f32(false, a, false, b,
                                                      (short)0, acc[nt],
                                                      false, false);
    }
  }

  const int mbase = mt * 16 + ((lane >> 4) << 3);
  #pragma unroll
  for (int nt = 0; nt < 4; ++nt) {
    const int n = n0 + nt * 16 + r;
    const float bv = bias[n];
    #pragma unroll
    for (int i = 0; i < 8; ++i)
      Y[(size_t)(mbase + i) * N + n] = acc[nt][i] + bv;
  }
}

// ---------------- K3: NAT neighborhood attention (3x3, 4 heads) --------------
__global__ __launch_bounds__(256) void k_nat_attn(const float* __restrict__ qkv,
                                                  const float* __restrict__ rpb,
                                                  float* __restrict__ natt) {
  int tid = blockIdx.x * blockDim.x + threadIdx.x;     // over B*NH*HW
  if (tid >= B_ * NH_ * HW_) return;
  int x = tid % W_;
  int y = (tid / W_) % H_;
  int h = (tid / HW_) % NH_;
  int b = tid / (HW_ * NH_);
  const size_t t0 = (size_t)(b * HW_ + y * W_ + x);
  const float* qr = qkv + t0 * 384 + h * HD_;          // q slab (s=0)
  float q[HD_];
  #pragma unroll
  for (int d = 0; d < HD_; ++d) q[d] = qr[d];

  const int ys = min(max(y - 1, 0), H_ - KW_);
  const int xs = min(max(x - 1, 0), W_ - KW_);
  const float scale = 0.17677669529663687f;            // HD^-0.5

  float logit[9];
  #pragma unroll
  for (int i = 0; i < KW_; ++i) {
    int yy = ys + i, rh = yy - y + (KW_ - 1);
    #pragma unroll
    for (int j = 0; j < KW_; ++j) {
      int xx = xs + j, rw = xx - x + (KW_ - 1);
      const float* kr = qkv + (size_t)(b * HW_ + yy * W_ + xx) * 384 + 128 + h * HD_;
      float s = 0.f;
      #pragma unroll
      for (int d = 0; d < HD_; ++d) s += q[d] * kr[d];
      logit[i * KW_ + j] = s * scale + rpb[h * 25 + rh * 5 + rw];
    }
  }
  float mx = logit[0];
  #pragma unroll
  for (int p = 1; p < 9; ++p) mx = fmaxf(mx, logit[p]);
  float Z = 0.f;
  #pragma unroll
  for (int p = 0; p < 9; ++p) { logit[p] = __expf(logit[p] - mx); Z += logit[p]; }
  float inv = 1.0f / Z;

  float acc[HD_];
  #pragma unroll
  for (int d = 0; d < HD_; ++d) acc[d] = 0.f;
  #pragma unroll
  for (int i = 0; i < KW_; ++i) {
    int yy = ys + i;
    #pragma unroll
    for (int j = 0; j < KW_; ++j) {
      int xx = xs + j;
      float p = logit[i * KW_ + j] * inv;
      const float* vr = qkv + (size_t)(b * HW_ + yy * W_ + xx) * 384 + 256 + h * HD_;
      #pragma unroll
      for (int d = 0; d < HD_; ++d) acc[d] += p * vr[d];
    }
  }
  float* outp = natt + t0 * C_ + h * HD_;
  #pragma unroll
  for (int d = 0; d < HD_; ++d) outp[d] = acc[d];
}

// ---------------- K6: DAT offset path (dwconv5x5 + LN + GELU + proj) ---------
__global__ __launch_bounds__(256) void k_dat_offsets(const float* __restrict__ q_l,
                                                     const float* __restrict__ dw_w,
                                                     const float* __restrict__ dw_b,
                                                     const float* __restrict__ ln_g,
                                                     const float* __restrict__ ln_b,
                                                     const float* __restrict__ off_w,
                                                     float* __restrict__ coords) {
  const int lane = threadIdx.x & 31;
  const int pid  = blockIdx.x * (blockDim.x >> 5) + (threadIdx.x >> 5);
  if (pid >= G_ * B_ * HW_) return;
  const int n  = pid % HW_;
  const int bg = pid / HW_;
  const int y = n / W_, x = n % W_;
  const int b = bg >> 1, g = bg & 1;

  float o[2];
  #pragma unroll
  for (int t = 0; t < 2; ++t) {
    const int cg = lane * 2 + t;
    const int c  = g * CG_ + cg;
    float acc = dw_b[cg];
    #pragma unroll
    for (int ky = 0; ky < KS_; ++ky) {
      int yy = y + ky - 2;
      if (yy < 0 || yy >= H_) continue;
      #pragma unroll
      for (int kx = 0; kx < KS_; ++kx) {
        int xx = x + kx - 2;
        if (xx < 0 || xx >= W_) continue;
        acc += q_l[(size_t)(b * HW_ + yy * W_ + xx) * C_ + c] *
               dw_w[cg * 25 + ky * 5 + kx];
      }
    }
    o[t] = acc;
  }
  float mean = wave_sum(o[0] + o[1]) * (1.0f / CG_);
  float var  = wave_sum(o[0] * o[0] + o[1] * o[1]) * (1.0f / CG_) - mean * mean;
  float rinv = rsqrtf(var + EPS_);
  float ge[2];
  #pragma unroll
  for (int t = 0; t < 2; ++t) {
    const int cg = lane * 2 + t;
    float v = (o[t] - mean) * rinv * ln_g[cg] + ln_b[cg];
    ge[t] = 0.5f * v * (1.0f + erff(v * 0.70710678118654752f));   // exact GELU
  }
  const int cg0 = lane * 2, cg1 = lane * 2 + 1;
  float off0 = wave_sum(ge[0] * off_w[cg0]        + ge[1] * off_w[cg1]);
  float off1 = wave_sum(ge[0] * off_w[CG_ + cg0]  + ge[1] * off_w[CG_ + cg1]);
  if (lane == 0) {
    const float rng = 3.0f;
    float offy = tanhf(off0) * (rng / (float)H_);
    float offx = tanhf(off1) * (rng / (float)W_);
    float ry = ((float)y + 0.5f) / (float)H_ * 2.0f - 1.0f;
    float rx = ((float)x + 0.5f) / (float)W_ * 2.0f - 1.0f;
    float posy = offy + ry, posx = offx + rx;
    coords[pid * 2 + 0] = (posx + 1.0f) * (float)(W_ - 1) * 0.5f;  // gx
    coords[pid * 2 + 1] = (posy + 1.0f) * (float)(H_ - 1) * 0.5f;  // gy
  }
}

// ---------------- K7: bilinear grid sample -----------------------------------
__global__ __launch_bounds__(256) void k_grid_sample(const float* __restrict__ y_l,
                                                     const float* __restrict__ coords,
                                                     float* __restrict__ xs_l) {
  int tid = blockIdx.x * blockDim.x + threadIdx.x;     // over (G*B*HW)*CG
  if (tid >= G_ * B_ * HW_ * CG_) return;
  const int cg  = tid & 63;
  const int pid = tid >> 6;
  const int n   = pid % HW_;
  const int bg  = pid / HW_;
  const int b = bg >> 1, g = bg & 1;
  const int c = g * CG_ + cg;

  const float gx = coords[pid * 2 + 0];
  const float gy = coords[pid * 2 + 1];
  const float x0f = floorf(gx), y0f = floorf(gy);
  const int x0 = (int)x0f, y0 = (int)y0f;
  const float fx = gx - x0f, fy = gy - y0f;

  float acc = 0.f;
  #pragma unroll
  for (int ty = 0; ty < 2; ++ty) {
    #pragma unroll
    for (int tx = 0; tx < 2; ++tx) {
      int xi = x0 + tx, yi = y0 + ty;
      float w = (tx ? fx : 1.0f - fx) * (ty ? fy : 1.0f - fy);
      bool valid = (xi >= 0) && (xi <= W_ - 1) && (yi >= 0) && (yi <= H_ - 1);
      int xc = min(max(xi, 0), W_ - 1);
      int yc = min(max(yi, 0), H_ - 1);
      float v = y_l[(size_t)(b * HW_ + yc * W_ + xc) * C_ + c];
      acc += v * (valid ? w : 0.0f);
    }
  }
  xs_l[(size_t)(b * HW_ + n) * C_ + c] = acc;
}

// ---------------- K10: DAT global attention, wave per (head-batch, row) ------
__global__ __launch_bounds__(256) void k_dat_attn(const float* __restrict__ q_l,
                                                  const float* __restrict__ k_l,
                                                  const float* __restrict__ v_l,
                                                  float* __restrict__ o_l) {
  const int lane = threadIdx.x & 31;
  const int gid  = blockIdx.x * (blockDim.x >> 5) + (threadIdx.x >> 5);
  if (gid >= B_ * NH_ * HW_) return;
  const int m  = gid % HW_;
  const int bh = gid / HW_;
  const int b = bh / NH_, h = bh % NH_;
  const float scale = 0.17677669529663687f;

  const float* qr = q_l + (size_t)(b * HW_ + m) * C_ + h * HD_;
  float q[HD_];
  #pragma unroll
  for (int d = 0; d < HD_; ++d) q[d] = qr[d];

  const int NC = (HW_ + 31) / 32;                      // 25 chunks
  float s[NC];
  #pragma unroll
  for (int c = 0; c < NC; ++c) {
    int n = c * 32 + lane;
    float a = -INFINITY;
    if (n < HW_) {
      const float* kr = k_l + (size_t)(b * HW_ + n) * C_ + h * HD_;
      float t = 0.f;
      #pragma unroll
      for (int d = 0; d < HD_; ++d) t += q[d] * kr[d];
      a = t * scale;
    }
    s[c] = a;
  }
  float mx = s[0];
  #pragma unroll
  for (int c = 1; c < NC; ++c) mx = fmaxf(mx, s[c]);
  mx = wave_max(mx);
  float Zl = 0.f;
  #pragma unroll
  for (int c = 0; c < NC; ++c) {
    float p = (s[c] == -INFINITY) ? 0.f : __expf(s[c] - mx);
    s[c] = p;
    Zl += p;
  }
  const float Zinv = 1.0f / wave_sum(Zl);

  float* outp = o_l + (size_t)(b * HW_ + m) * C_ + h * HD_;
  for (int d = 0; d < HD_; ++d) {
    float pd = 0.f;
    #pragma unroll
    for (int c = 0; c < NC; ++c) {
      int n = c * 32 + lane;
      if (n < HW_)
        pd += s[c] * v_l[(size_t)(b * HW_ + n) * C_ + h * HD_ + d];
    }
    pd = wave_sum(pd);
    if (lane == 0) outp[d] = pd * Zinv;
  }
}

// ---------------- K12: residual + relu ---------------------------------------
__global__ __launch_bounds__(256) void k_res_relu(const float* __restrict__ xl,
                                                  const float* __restrict__ z_l,
                                                  float* __restrict__ yf) {
  int tid = blockIdx.x * blockDim.x + threadIdx.x;
  if (tid >= T_ * C_) return;
  yf[tid] = fmaxf(xl[tid] + z_l[tid], 0.0f);
}

// ---------------- K13: downsample conv 3x3/s2 as implicit-im2col WMMA GEMM ---
__global__ __launch_bounds__(256) void k_down_conv_wmma(const float* __restrict__ yf,
                                                        const float* __restrict__ Wd,
                                                        float* __restrict__ cv) {
  const int lane = threadIdx.x & 31;
  const int wave = blockIdx.x * (blockDim.x >> 5) + (threadIdx.x >> 5);
  const int total = (B_ * OP_ / 16) * (OC_ / 16);      // 196*16 = 3136 waves
  if (wave >= total) return;
  const int mt = wave >> 4;
  const int nt = wave & 15;
  const int r  = lane & 15;
  const int kb = (lane >> 4) << 1;
  const int p  = mt * 16 + r;                          // output pixel 0..3135
  const int b  = p / OP_;
  const int loc = p % OP_;
  const int oy = loc / OW_, ox = loc % OW_;
  const float* wrow = Wd + (size_t)(nt * 16 + r) * KCONV_;

  v8f acc = {};
  for (int k = 0; k < KCONV_; k += 4) {
    const int kk0 = k + kb;
    v2f a;
    #pragma unroll
    for (int j = 0; j < 2; ++j) {
      int kk = kk0 + j;
      int c = kk / 9, kidx = kk % 9;
      int ky = kidx / 3, kx = kidx % 3;
      int yy = oy * 2 + ky - 1, xx = ox * 2 + kx - 1;
      float v = 0.f;
      if (yy >= 0 && yy < H_ && xx >= 0 && xx < W_)
        v = yf[(size_t)(b * HW_ + yy * W_ + xx) * C_ + c];
      a[j] = v;
    }
    v2f bb = *(const v2f*)(wrow + kk0);
    acc = __builtin_amdgcn_wmma_f32_16x16x4_f32(false, a, false, bb,
                                                (short)0, acc, false, false);
  }
  const int n = nt * 16 + r;
  const int mbase = mt * 16 + ((lane >> 4) << 3);
  #pragma unroll
  for (int i = 0; i < 8; ++i)
    cv[(size_t)(mbase + i) * OC_ + n] = acc[i];
}

// ---------------- K14: channel layernorm(256) + write NCHW output ------------
__global__ __launch_bounds__(256) void k_final_ln(const float* __restrict__ cv,
                                                  const float* __restrict__ g,
                                                  const float* __restrict__ be,
                                                  float* __restrict__ out) {
  const int lane = threadIdx.x & 31;
  const int p    = blockIdx.x * (blockDim.x >> 5) + (threadIdx.x >> 5);
  if (p >= B_ * OP_) return;
  const int b = p / OP_, loc = p % OP_;
  const float* row = cv + (size_t)p * OC_;
  float vals[8];
  float s = 0.f, sq = 0.f;
  #pragma unroll
  for (int j = 0; j < 8; ++j) {
    float v = row[lane + 32 * j];
    vals[j] = v;
    s += v; sq += v * v;
  }
  float mean = wave_sum(s) * (1.0f / OC_);
  float var  = wave_sum(sq) * (1.0f / OC_) - mean * mean;
  float rinv = rsqrtf(var + EPS_);
  #pragma unroll
  for (int j = 0; j < 8; ++j) {
    int c = lane + 32 * j;
    out[(size_t)b * OC_ * OP_ + (size_t)c * OP_ + loc] =
        (vals[j] - mean) * rinv * g[c] + be[c];
  }
}

extern "C" void kernel_launch(void* const* d_in, const int* in_sizes, int n_in,
                              void* d_out, int out_size, void* d_ws, size_t ws_size,
                              hipStream_t stream) {
  const float* x        = (const float*)d_in[0];
  const float* qkv_w    = (const float*)d_in[1];
  const float* qkv_b    = (const float*)d_in[2];
  const float* rpb      = (const float*)d_in[3];
  const float* proj_w   = (const float*)d_in[4];
  const float* proj_b   = (const float*)d_in[5];
  const float* pq_w     = (const float*)d_in[6];
  const float* pq_b     = (const float*)d_in[7];
  const float* pk_w     = (const float*)d_in[8];
  const float* pk_b     = (const float*)d_in[9];
  const float* pv_w     = (const float*)d_in[10];
  const float* pv_b     = (const float*)d_in[11];
  const float* po_w     = (const float*)d_in[12];
  const float* po_b     = (const float*)d_in[13];
  const float* dw_w     = (const float*)d_in[14];
  const float* dw_b     = (const float*)d_in[15];
  const float* ln_g     = (const float*)d_in[16];
  const float* ln_b     = (const float*)d_in[17];
  const float* off_w    = (const float*)d_in[18];
  const float* down_w   = (const float*)d_in[19];
  const float* down_g   = (const float*)d_in[20];
  const float* down_b   = (const float*)d_in[21];
  float* out = (float*)d_out;

  const size_t TC = (size_t)T_ * C_;                   // 1,605,632
  float* ws     = (float*)d_ws;
  float* xl     = ws;                                  // TC   (persists to residual)
  float* qkv    = ws + TC;                             // 3*TC (reused: xs,k,v)
  float* natt   = ws + 4 * TC;                         // TC   (reused: o_l)
  float* y_l    = ws + 5 * TC;                         // TC   (reused: z_l)
  float* q_l    = ws + 6 * TC;                         // TC   (reused: yf)
  float* coords = ws + 7 * TC;                         // 50,176
  float* cv     = ws + 7 * TC + (size_t)G_ * B_ * HW_ * 2;  // 802,816
  float* xs_l = qkv;
  float* k_l  = qkv + TC;
  float* v_l  = qkv + 2 * TC;
  float* o_l  = natt;
  float* z_l  = y_l;
  float* yf   = q_l;

  const dim3 blk(256);
  const dim3 g128(98, 2);                              // 784/8 m-blocks x 128/64
  const dim3 g384(98, 6);                              // 784/8 m-blocks x 384/64

  // K1: layout transform
  k_nchw_to_tok<<<(B_ * C_ * HW_ + 255) / 256, blk, 0, stream>>>(x, xl);
  // K2: qkv GEMM  T x 384 x 128
  k_gemm_wmma_lds<<<g384, blk, 0, stream>>>(xl, qkv_w, qkv_b, qkv, 384);
  // K3: NAT neighborhood attention
  k_nat_attn<<<(B_ * NH_ * HW_ + 255) / 256, blk, 0, stream>>>(qkv, rpb, natt);
  // K4: NAT projection
  k_gemm_wmma_lds<<<g128, blk, 0, stream>>>(natt, proj_w, proj_b, y_l, C_);
  // K5: DAT q projection
  k_gemm_wmma_lds<<<g128, blk, 0, stream>>>(y_l, pq_w, pq_b, q_l, C_);
  // K6: DAT offsets (dwconv + LN + GELU + proj + tanh) -> sample coords
  k_dat_offsets<<<(G_ * B_ * HW_ + 7) / 8, blk, 0, stream>>>(q_l, dw_w, dw_b, ln_g, ln_b,
                                                             off_w, coords);
  // K7: bilinear grid sample from y (NAT output)
  k_grid_sample<<<(G_ * B_ * HW_ * CG_ + 255) / 256, blk, 0, stream>>>(y_l, coords, xs_l);
  // K8/K9: k,v projections on sampled features
  k_gemm_wmma_lds<<<g128, blk, 0, stream>>>(xs_l, pk_w, pk_b, k_l, C_);
  k_gemm_wmma_lds<<<g128, blk, 0, stream>>>(xs_l, pv_w, pv_b, v_l, C_);
  // K10: global attention, fused softmax (wave per row)
  k_dat_attn<<<(B_ * NH_ * HW_ + 7) / 8, blk, 0, stream>>>(q_l, k_l, v_l, o_l);
  // K11: output projection
  k_gemm_wmma_lds<<<g128, blk, 0, stream>>>(o_l, po_w, po_b, z_l, C_);
  // K12: residual + relu
  k_res_relu<<<(T_ * C_ + 255) / 256, blk, 0, stream>>>(xl, z_l, yf);
  // K13: downsample conv via implicit-im2col WMMA GEMM
  k_down_conv_wmma<<<(3136 + 7) / 8, blk, 0, stream>>>(yf, down_w, cv);
  // K14: final channel layernorm, write NCHW output
  k_final_ln<<<(B_ * OP_ + 7) / 8, blk, 0, stream>>>(cv, down_g, down_b, out);
}